// CenterLoss_3917010174521
// MI455X (gfx1250) — compile-verified
//
#include <hip/hip_runtime.h>

typedef __attribute__((ext_vector_type(2))) float v2f;
typedef __attribute__((ext_vector_type(8))) float v8f;

#define CL_MIN 1e-12f
#define CL_MAX 1e12f

__device__ __forceinline__ float readlane_f(float v, int lane) {
    return __builtin_bit_cast(
        float, __builtin_amdgcn_readlane(__builtin_bit_cast(int, v), lane));
}

__global__ void center_loss_zero(float* out) { out[0] = 0.0f; }

// One wave (32 lanes) owns a 16-row tile. For each K-chunk of 4 columns it
// builds A = (x - c) in the V_WMMA_F32_16X16X4_F32 A-layout (lane = row M,
// VGPR/half = K) and accumulates acc += A * A^T. With the mirrored f32 B
// layout, A^T occupies the same registers as A, so src0 == src1. The diagonal
// of the accumulated 16x16 Gram matrix is ||x_m - c_m||^2 per row.
__global__ void __launch_bounds__(256)
center_loss_wmma(const float* __restrict__ x,
                 const long long* __restrict__ labels,
                 const float* __restrict__ centers,
                 float* __restrict__ out,
                 int D, int B, float inv_B) {
    const int lane = threadIdx.x & 31;
    const int wave = threadIdx.x >> 5;
    const int tile = blockIdx.x * 8 + wave;   // 16-row tile index
    const int row_base = tile * 16;
    if (row_base >= B) return;                // uniform per wave: EXEC stays all-1s

    const int r    = lane & 15;               // row within tile (M)
    const int half = lane >> 4;               // 0 -> K 0..1, 1 -> K 2..3
    const int row  = row_base + r;

    const int lbl = (int)labels[row];         // int64 labels, value < 751
    const float* __restrict__ xrow = x + (size_t)row * (size_t)D;
    const float* __restrict__ crow = centers + (size_t)lbl * (size_t)D;
    const int koff = 2 * half;

    v8f acc = {};
    for (int k0 = 0; k0 < D; k0 += 32) {
        // keep the unique x stream ahead of the WMMA pipe (centers sit in L2)
        __builtin_prefetch(xrow + k0 + 32, 0, 0);
#pragma unroll
        for (int kk = 0; kk < 32; kk += 4) {
            const int k = k0 + kk + koff;
            v2f xv = *(const v2f*)(xrow + k);
            v2f cv = *(const v2f*)(crow + k);
            v2f a  = xv - cv;
            // 8-arg form: (neg_a, A, neg_b, B, c_mod, C, reuse_a, reuse_b)
            acc = __builtin_amdgcn_wmma_f32_16x16x4_f32(
                false, a, false, a, (short)0, acc, false, false);
        }
    }

    // Diagonal extraction: D[m][m] -> (VGPR m, lane m); D[m+8][m+8] -> (VGPR m, lane m+24)
    float tsum = 0.0f;
#pragma unroll
    for (int m = 0; m < 8; ++m) {
        float d0 = readlane_f(acc[m], m);
        float d1 = readlane_f(acc[m], m + 24);
        d0 = fminf(fmaxf(d0, CL_MIN), CL_MAX);
        d1 = fminf(fmaxf(d1, CL_MIN), CL_MAX);
        tsum += d0 + d1;
    }
    if (lane == 0) atomicAdd(out, tsum * inv_B);
}

extern "C" void kernel_launch(void* const* d_in, const int* in_sizes, int n_in,
                              void* d_out, int out_size, void* d_ws, size_t ws_size,
                              hipStream_t stream) {
    const float*     xp = (const float*)d_in[0];
    const long long* lp = (const long long*)d_in[1];   // int64 labels
    const float*     cp = (const float*)d_in[2];
    float*           op = (float*)d_out;

    const int B = in_sizes[1];                 // 16384
    const int D = in_sizes[0] / B;             // 2048
    const int tiles  = B / 16;                 // 1024
    const int blocks = (tiles + 7) / 8;        // 8 waves (tiles) per 256-thread block

    center_loss_zero<<<1, 1, 0, stream>>>(op);
    center_loss_wmma<<<blocks, 256, 0, stream>>>(xp, lp, cp, op, D, B, 1.0f / (float)B);
}